// GloAttnConv_23132693856311
// MI455X (gfx1250) — compile-verified
//
#include <hip/hip_runtime.h>

typedef __attribute__((ext_vector_type(2))) float v2f;
typedef __attribute__((ext_vector_type(8))) float v8f;

#define WMMA_F32(A_, B_, C_) \
  __builtin_amdgcn_wmma_f32_16x16x4_f32(false, (A_), false, (B_), (short)0, (C_), false, false)

// Problem constants
// N=131072, B=64, NPG=2048, C_IN=64, H=4, D=64, HD=256

// Workspace layout (float offsets). Total = 5,345,280 floats = 21.4 MB.
constexpr size_t OFF_PKVS  = 0;                              // [256][4][64][64] partial kvs per (b,chunk)
constexpr size_t OFF_PKSUM = OFF_PKVS  + (size_t)256*4*64*64; // [256][4][64]
constexpr size_t OFF_PXSUM = OFF_PKSUM + (size_t)256*256;     // [256][64]
constexpr size_t OFF_FKVS  = OFF_PXSUM + (size_t)256*64;      // [64][4][64][64]
constexpr size_t OFF_FKSUM = OFF_FKVS  + (size_t)64*4*64*64;  // [64][4][64]
constexpr size_t OFF_FXSUM = OFF_FKSUM + (size_t)64*256;      // [64][64]

// ---------------------------------------------------------------------------
// Pass 1: per (graph b, 512-row chunk): k = normalize(x @ Wk^T + bk) per head,
// accumulate kvs[h] += k^T x, ksum[h] += colsum(k), xsum += colsum(x).
// grid = 256 (b*4 + chunk), block = 256 (8 waves).
// ---------------------------------------------------------------------------
__global__ __launch_bounds__(256) void k_pass1(const float* __restrict__ x,
                                               const float* __restrict__ Wk,
                                               const float* __restrict__ Wkb,
                                               float* __restrict__ ws)
{
  __shared__ float xs[512 * 64];        // 128 KB : x chunk
  __shared__ float kls[32 * 256];       // 32 KB  : normalized k for 2 row-tiles (all heads)
  __shared__ float ksum_lds[8 * 64];    // 2 KB

  const int blk  = blockIdx.x;          // = b*4 + chunk
  const int row0 = blk * 512;           // global node row
  const int tid  = threadIdx.x;
  const int wave = tid >> 5;
  const int lane = tid & 31;
  const int lo   = lane & 15;
  const int hi   = lane >> 4;

  // Cooperative load of x chunk (512x64 f32) into LDS.
  {
    const float4* src = (const float4*)(x + (size_t)row0 * 64);
    float4*       dst = (float4*)xs;
    for (int i = tid; i < 512 * 64 / 4; i += 256) dst[i] = src[i];
  }
  __syncthreads();

  const int hA  = wave & 3;             // head handled in stage A
  const int rtl = wave >> 2;            // row-tile (0/1) within 32-row group

  v8f   acc2[2][4] = {};                // persistent kvs accumulators: 2 (h,dt) combos x 4 c-tiles
  float kp[4] = {0.f, 0.f, 0.f, 0.f};   // ksum partials (stage A head)

  for (int it = 0; it < 16; ++it) {     // 32 rows per iteration
    // ---- Stage A: k projection + bias + row-normalize for (head hA, row-tile) ----
    const int rowBase = it * 32 + rtl * 16;
    v8f acc[4] = {};
    for (int ks = 0; ks < 16; ++ks) {
      v2f a = *(const v2f*)&xs[(rowBase + lo) * 64 + ks * 4 + 2 * hi];
      for (int ct = 0; ct < 4; ++ct) {
        v2f bb = *(const v2f*)&Wk[(size_t)(hA * 64 + ct * 16 + lo) * 64 + ks * 4 + 2 * hi];
        acc[ct] = WMMA_F32(a, bb, acc[ct]);
      }
    }
    for (int ct = 0; ct < 4; ++ct) {
      float bias = Wkb[hA * 64 + ct * 16 + lo];
      for (int v = 0; v < 8; ++v) acc[ct][v] += bias;
    }
    for (int v = 0; v < 8; ++v) {       // row L2-normalize (rows v+8*hi)
      float s = 0.f;
      for (int ct = 0; ct < 4; ++ct) s += acc[ct][v] * acc[ct][v];
      s += __shfl_xor(s, 1, 32); s += __shfl_xor(s, 2, 32);
      s += __shfl_xor(s, 4, 32); s += __shfl_xor(s, 8, 32);
      float r = 1.0f / sqrtf(s);
      for (int ct = 0; ct < 4; ++ct) acc[ct][v] *= r;
    }
    for (int ct = 0; ct < 4; ++ct) {    // ksum partial + spill k tile for transpose
      float cs = 0.f;
      for (int v = 0; v < 8; ++v) {
        cs += acc[ct][v];
        kls[(rtl * 16 + v + 8 * hi) * 256 + hA * 64 + ct * 16 + lo] = acc[ct][v];
      }
      kp[ct] += cs;
    }
    __syncthreads();

    // ---- Stage B: kvs[h][d][c] += k^T x over these 32 rows ----
    for (int j = 0; j < 2; ++j) {
      const int combo = wave * 2 + j;
      const int h2 = combo >> 2, dt = combo & 3;
      for (int ks = 0; ks < 8; ++ks) {
        const int n0 = ks * 4 + 2 * hi;
        v2f a;
        a.x = kls[n0 * 256 + h2 * 64 + dt * 16 + lo];
        a.y = kls[(n0 + 1) * 256 + h2 * 64 + dt * 16 + lo];
        for (int ct = 0; ct < 4; ++ct) {
          v2f bb;
          bb.x = xs[(it * 32 + n0) * 64 + ct * 16 + lo];
          bb.y = xs[(it * 32 + n0 + 1) * 64 + ct * 16 + lo];
          acc2[j][ct] = WMMA_F32(a, bb, acc2[j][ct]);
        }
      }
    }
    __syncthreads();
  }

  // Write kvs partials [h][d][c]
  float* pkvs = ws + OFF_PKVS + (size_t)blk * 16384;
  for (int j = 0; j < 2; ++j) {
    const int combo = wave * 2 + j, h2 = combo >> 2, dt = combo & 3;
    for (int ct = 0; ct < 4; ++ct)
      for (int v = 0; v < 8; ++v)
        pkvs[h2 * 4096 + (dt * 16 + v + 8 * hi) * 64 + ct * 16 + lo] = acc2[j][ct][v];
  }
  // ksum: fold halves, then the two waves sharing each head
  for (int ct = 0; ct < 4; ++ct) {
    kp[ct] += __shfl_xor(kp[ct], 16, 32);
    if (hi == 0) ksum_lds[wave * 64 + ct * 16 + lo] = kp[ct];
  }
  __syncthreads();
  {
    int h = tid >> 6, d = tid & 63;
    ws[OFF_PKSUM + (size_t)blk * 256 + tid] =
        ksum_lds[h * 64 + d] + ksum_lds[(h + 4) * 64 + d];
  }
  if (tid < 64) {
    float s = 0.f;
    for (int r = 0; r < 512; ++r) s += xs[r * 64 + tid];
    ws[OFF_PXSUM + (size_t)blk * 64 + tid] = s;
  }
}

// ---------------------------------------------------------------------------
// Pass 1.5: reduce 4 chunk partials -> per-graph kvs/ksum/xsum. grid=64.
// ---------------------------------------------------------------------------
__global__ __launch_bounds__(256) void k_reduce(float* __restrict__ ws)
{
  const int b = blockIdx.x, tid = threadIdx.x;
  for (int idx = tid; idx < 16384; idx += 256) {
    float s = 0.f;
    for (int c = 0; c < 4; ++c) s += ws[OFF_PKVS + (size_t)(b * 4 + c) * 16384 + idx];
    ws[OFF_FKVS + (size_t)b * 16384 + idx] = s;
  }
  {
    float s = 0.f;
    for (int c = 0; c < 4; ++c) s += ws[OFF_PKSUM + (size_t)(b * 4 + c) * 256 + tid];
    ws[OFF_FKSUM + (size_t)b * 256 + tid] = s;
  }
  if (tid < 64) {
    float s = 0.f;
    for (int c = 0; c < 4; ++c) s += ws[OFF_PXSUM + (size_t)(b * 4 + c) * 64 + tid];
    ws[OFF_FXSUM + (size_t)b * 64 + tid] = s;
  }
}

// ---------------------------------------------------------------------------
// Pass 2: q = normalize(x @ Wq^T + bq); out = (q@kvs + xsum) / (q.ksum + n).
// grid = 512 (b*8 + chunk of 256 rows), block = 256 (8 waves).
// ---------------------------------------------------------------------------
__global__ __launch_bounds__(256) void k_pass2(const float* __restrict__ x,
                                               const float* __restrict__ Wq,
                                               const float* __restrict__ Wqb,
                                               const int*   __restrict__ n_nodes,
                                               const float* __restrict__ ws,
                                               float* __restrict__ out)
{
  __shared__ float xs[256 * 64];        // 64 KB
  __shared__ float kvs[4 * 64 * 64];    // 64 KB
  __shared__ float qls[8 * 16 * 64];    // 32 KB wave-private q transpose scratch
  __shared__ float ksum_l[256];
  __shared__ float xsum_l[64];

  const int blk  = blockIdx.x;          // = b*8 + chunk
  const int b    = blk >> 3;
  const int row0 = blk * 256;           // global node row
  const int tid  = threadIdx.x;
  const int wave = tid >> 5, lane = tid & 31, lo = lane & 15, hi = lane >> 4;

  {
    const float4* src = (const float4*)(x + (size_t)row0 * 64);
    float4*       dst = (float4*)xs;
    for (int i = tid; i < 256 * 64 / 4; i += 256) dst[i] = src[i];
    const float4* ks = (const float4*)(ws + OFF_FKVS + (size_t)b * 16384);
    float4*       kd = (float4*)kvs;
    for (int i = tid; i < 16384 / 4; i += 256) kd[i] = ks[i];
    ksum_l[tid] = ws[OFF_FKSUM + (size_t)b * 256 + tid];
    if (tid < 64) xsum_l[tid] = ws[OFF_FXSUM + (size_t)b * 64 + tid];
  }
  const float ng = (float)n_nodes[b];
  __syncthreads();

  const int h   = wave & 3;
  const int rtl = wave >> 2;
  float* qw = qls + wave * 1024;        // 16x64 wave-private tile

  for (int it = 0; it < 8; ++it) {
    const int rowBase = it * 32 + rtl * 16;

    // q projection (16x64) via WMMA
    v8f acc[4] = {};
    for (int ks = 0; ks < 16; ++ks) {
      v2f a = *(const v2f*)&xs[(rowBase + lo) * 64 + ks * 4 + 2 * hi];
      for (int ct = 0; ct < 4; ++ct) {
        v2f bb = *(const v2f*)&Wq[(size_t)(h * 64 + ct * 16 + lo) * 64 + ks * 4 + 2 * hi];
        acc[ct] = WMMA_F32(a, bb, acc[ct]);
      }
    }
    for (int ct = 0; ct < 4; ++ct) {
      float bias = Wqb[h * 64 + ct * 16 + lo];
      for (int v = 0; v < 8; ++v) acc[ct][v] += bias;
    }

    // normalize rows + norm denominator q.ksum + n
    float normv[8];
    for (int v = 0; v < 8; ++v) {
      float s = 0.f;
      for (int ct = 0; ct < 4; ++ct) s += acc[ct][v] * acc[ct][v];
      s += __shfl_xor(s, 1, 32); s += __shfl_xor(s, 2, 32);
      s += __shfl_xor(s, 4, 32); s += __shfl_xor(s, 8, 32);
      float r = 1.0f / sqrtf(s);
      for (int ct = 0; ct < 4; ++ct) acc[ct][v] *= r;
      float t = 0.f;
      for (int ct = 0; ct < 4; ++ct) t += acc[ct][v] * ksum_l[h * 64 + ct * 16 + lo];
      t += __shfl_xor(t, 1, 32); t += __shfl_xor(t, 2, 32);
      t += __shfl_xor(t, 4, 32); t += __shfl_xor(t, 8, 32);
      normv[v] = t + ng;
    }

    // transpose q via wave-private LDS (same-wave LDS is in-order)
    for (int ct = 0; ct < 4; ++ct)
      for (int v = 0; v < 8; ++v)
        qw[(v + 8 * hi) * 64 + ct * 16 + lo] = acc[ct][v];

    // num = q @ kvs[h]
    v8f nacc[4] = {};
    for (int ks = 0; ks < 16; ++ks) {
      v2f a = *(const v2f*)&qw[lo * 64 + ks * 4 + 2 * hi];
      for (int ct = 0; ct < 4; ++ct) {
        v2f bb;
        bb.x = kvs[h * 4096 + (ks * 4 + 2 * hi) * 64 + ct * 16 + lo];
        bb.y = kvs[h * 4096 + (ks * 4 + 2 * hi + 1) * 64 + ct * 16 + lo];
        nacc[ct] = WMMA_F32(a, bb, nacc[ct]);
      }
    }

    // epilogue: (num + xsum) / norm -> out[node][h*64 + c]
    for (int ct = 0; ct < 4; ++ct) {
      float xv = xsum_l[ct * 16 + lo];
      for (int v = 0; v < 8; ++v) {
        float o = (nacc[ct][v] + xv) / normv[v];
        out[(size_t)(row0 + rowBase + v + 8 * hi) * 256 + h * 64 + ct * 16 + lo] = o;
      }
    }
  }
}

// ---------------------------------------------------------------------------
extern "C" void kernel_launch(void* const* d_in, const int* in_sizes, int n_in,
                              void* d_out, int out_size, void* d_ws, size_t ws_size,
                              hipStream_t stream)
{
  const float* x    = (const float*)d_in[0];
  const float* Wq_w = (const float*)d_in[1];
  const float* Wq_b = (const float*)d_in[2];
  const float* Wk_w = (const float*)d_in[3];
  const float* Wk_b = (const float*)d_in[4];
  const int*   nn   = (const int*)d_in[5];
  float* ws  = (float*)d_ws;
  float* out = (float*)d_out;

  k_pass1 <<<256, 256, 0, stream>>>(x, Wk_w, Wk_b, ws);
  k_reduce<<< 64, 256, 0, stream>>>(ws);
  k_pass2 <<<512, 256, 0, stream>>>(x, Wq_w, Wq_b, nn, ws, out);
}